// newIF_18193481466346
// MI455X (gfx1250) — compile-verified
//
#include <hip/hip_runtime.h>
#include <hip/hip_bf16.h>
#include <stdint.h>

#define T_STEPS 8
#define BLOCK_THREADS 256
#define VEC 4                          // floats per thread per timestep (b128)
#define NEURONS_PER_BLOCK (BLOCK_THREADS * VEC)

typedef float v4f __attribute__((ext_vector_type(4)));
typedef int v4i __attribute__((vector_size(16)));  // matches builtin's pointee type

// ---------------- CDNA5 async global->LDS load (ASYNCcnt-tracked) ----------------
__device__ __forceinline__ void async_load_b128(const float* __restrict__ gsrc,
                                                v4f* lds_dst) {
#if defined(__HIP_DEVICE_COMPILE__)
#if __has_builtin(__builtin_amdgcn_global_load_async_to_lds_b128)
  __builtin_amdgcn_global_load_async_to_lds_b128(
      (__attribute__((address_space(1))) v4i*)(void*)gsrc,
      (__attribute__((address_space(3))) v4i*)(void*)lds_dst,
      /*imm offset*/ 0, /*cpol*/ 0);
#else
  unsigned lds_off =
      (unsigned)(uintptr_t)(__attribute__((address_space(3))) void*)(void*)lds_dst;
  asm volatile("global_load_async_to_lds_b128 %0, %1, off"
               :
               : "v"(lds_off), "v"(gsrc)
               : "memory");
#endif
#endif
}

#if defined(__HIP_DEVICE_COMPILE__)
#if __has_builtin(__builtin_amdgcn_s_wait_asynccnt)
#define WAIT_ASYNC(n) __builtin_amdgcn_s_wait_asynccnt(n)
#else
#define WAIT_ASYNC(n) asm volatile("s_wait_asynccnt %0" ::"n"(n) : "memory")
#endif
#else
#define WAIT_ASYNC(n) ((void)0)
#endif

// ---------------- staged consumption: wait asynccnt <= (T-1-t) before slice t ----
template <int t>
struct StepT {
  static __device__ __forceinline__ void run(const v4f* __restrict__ buf, int tid,
                                             float thre, float mem[VEC],
                                             float cnt[VEC], unsigned msk[VEC]) {
    WAIT_ASYNC(T_STEPS - 1 - t);  // loads complete in order; slice t has landed
    v4f xv = buf[t * BLOCK_THREADS + tid];
#pragma unroll
    for (int j = 0; j < VEC; ++j) {
      mem[j] += xv[j];
      const bool sp = (mem[j] >= thre);     // zif(mem - thre) forward
      msk[j] |= sp ? (1u << t) : 0u;
      cnt[j] += sp ? 1.0f : 0.0f;
      mem[j] -= sp ? thre : 0.0f;           // soft reset
    }
    StepT<t + 1>::run(buf, tid, thre, mem, cnt, msk);
  }
};
template <>
struct StepT<T_STEPS> {
  static __device__ __forceinline__ void run(const v4f*, int, float, float*, float*,
                                             unsigned*) {}
};

__global__ __launch_bounds__(BLOCK_THREADS) void snn_if_kernel(
    const float* __restrict__ x, const float* __restrict__ thresh,
    float* __restrict__ out, int N) {
  __shared__ v4f buf[T_STEPS * BLOCK_THREADS];  // 32 KB: 8 time-slices, b128/lane

  const int tid = threadIdx.x;
  const long long n0 =
      (long long)blockIdx.x * NEURONS_PER_BLOCK + (long long)tid * VEC;
  if (n0 >= (long long)N) return;

  // Issue all 8 ASYNCcnt-tracked streaming loads up front (deep MLP, no VGPR cost)
#pragma unroll
  for (int t = 0; t < T_STEPS; ++t) {
    async_load_b128(x + (long long)t * N + n0, &buf[t * BLOCK_THREADS + tid]);
  }

  const float thre = thresh[0];  // uniform scalar (s_load), overlaps async waits

  float mem[VEC], cnt[VEC];
  unsigned msk[VEC];
#pragma unroll
  for (int j = 0; j < VEC; ++j) {
    mem[j] = 0.5f * thre;
    cnt[j] = 0.0f;
    msk[j] = 0u;
  }

  // Each lane consumes only the LDS bytes its own async loads wrote -> no barrier.
  StepT<0>::run(buf, tid, thre, mem, cnt, msk);

  // Per-neuron effective threshold (final-step compensation)
  v4f nthre;
#pragma unroll
  for (int j = 0; j < VEC; ++j) {
    float compen = (mem[j] - 0.5f * thre) + cnt[j] * thre;
    compen = fminf(compen, (float)T_STEPS * thre);
    const bool cond = (compen > 0.0f) && (cnt[j] > 0.0f);
    nthre[j] = cond ? (compen / cnt[j]) : 0.0f;
  }

  // Stream out: spikes * new_thre, non-temporal b128 (write-once, bypass L2 residency)
#pragma unroll
  for (int t = 0; t < T_STEPS; ++t) {
    v4f o;
#pragma unroll
    for (int j = 0; j < VEC; ++j) {
      o[j] = ((msk[j] >> t) & 1u) ? nthre[j] : 0.0f;
    }
    __builtin_nontemporal_store(o, (v4f*)(out + (long long)t * N + n0));
  }
}

extern "C" void kernel_launch(void* const* d_in, const int* in_sizes, int n_in,
                              void* d_out, int out_size, void* d_ws, size_t ws_size,
                              hipStream_t stream) {
  (void)n_in;
  (void)out_size;
  (void)d_ws;
  (void)ws_size;
  const float* x = (const float*)d_in[0];       // [T*B, C, H, W] fp32
  const float* thresh = (const float*)d_in[1];  // [1] fp32
  float* out = (float*)d_out;

  const int total = in_sizes[0];
  const int N = total / T_STEPS;  // neurons = B*C*H*W

  const int grid = (N + NEURONS_PER_BLOCK - 1) / NEURONS_PER_BLOCK;
  snn_if_kernel<<<grid, BLOCK_THREADS, 0, stream>>>(x, thresh, out, N);
}